// AngularLayer_18545668784562
// MI455X (gfx1250) — compile-verified
//
#include <hip/hip_runtime.h>
#include <cstdint>
#include <cstddef>

#define ROWS     16      // rows per workgroup tile
#define NT       256     // 8 wave32s
#define IN_COLS  63
#define OUT_COLS 483
#define NPAIRS   210

// Cache policy immediates (ISA §7.3): TH in cpol[2:0]; TH_NT = 1.
// Both streams are touch-once (1.1 GB through a 192 MB L2) -> non-temporal.
#define CPOL_NT  1

// ---------------------------------------------------------------------------
// Tensor Data Mover (CDNA5) descriptor helpers — see CDNA5 ISA
// ---------------------------------------------------------------------------
#if defined(__HIP_DEVICE_COMPILE__) && \
    __has_builtin(__builtin_amdgcn_tensor_load_to_lds) && \
    __has_builtin(__builtin_amdgcn_tensor_store_from_lds) && \
    __has_builtin(__builtin_amdgcn_s_wait_tensorcnt)
#define USE_TDM 1
#else
#define USE_TDM 0
#endif

#if USE_TDM
typedef unsigned int v4u __attribute__((ext_vector_type(4)));
typedef int          v8i __attribute__((ext_vector_type(8)));
typedef int          v4i __attribute__((ext_vector_type(4)));

// Pack a 2-D f32 tile descriptor:
//  group0: [1:0]=count=1, lds_addr, global_addr(57b), type=2 at [127:126]
//  group1: data_size=4B(code 2), tensor_dim0/1, tile_dim0/1, tensor_dim0_stride
__device__ __forceinline__ void tdm_build_2d_f32(
    v4u& g0, v8i& g1,
    unsigned lds_byte_off, unsigned long long gaddr,
    unsigned tensor_d0, unsigned tensor_d1,
    unsigned tile_d0, unsigned tile_d1, unsigned stride0)
{
  g0.x = 1u;                                                  // count=1, user mode
  g0.y = lds_byte_off;                                        // lds_addr (bytes)
  g0.z = (unsigned)(gaddr & 0xFFFFFFFFull);                   // global_addr[31:0]
  g0.w = (unsigned)((gaddr >> 32) & 0x1FFFFFFull)             // global_addr[56:32]
       | (2u << 30);                                          // type = 2 ("image")
  g1[0] = (int)(2u << 16);                                    // wg_mask=0, data_size=4B
  g1[1] = (int)((tensor_d0 & 0xFFFFu) << 16);                 // tensor_dim0[15:0]
  g1[2] = (int)(((tensor_d0 >> 16) & 0xFFFFu)                 // tensor_dim0[31:16]
              | ((tensor_d1 & 0xFFFFu) << 16));               // tensor_dim1[15:0]
  g1[3] = (int)(((tensor_d1 >> 16) & 0xFFFFu)                 // tensor_dim1[31:16]
              | ((tile_d0 & 0xFFFFu) << 16));                 // tile_dim0
  g1[4] = (int)(tile_d1 & 0xFFFFu);                           // tile_dim1, tile_dim2=0
  g1[5] = (int)stride0;                                       // tensor_dim0_stride[31:0]
  g1[6] = 0;                                                  // stride hi / dim1_stride lo
  g1[7] = 0;
}

__device__ __forceinline__ void tdm_load_2d_f32(
    unsigned lds_byte_off, unsigned long long gaddr,
    unsigned tensor_d0, unsigned tensor_d1,
    unsigned tile_d0, unsigned tile_d1, unsigned stride0)
{
  v4u g0; v8i g1;
  tdm_build_2d_f32(g0, g1, lds_byte_off, gaddr, tensor_d0, tensor_d1,
                   tile_d0, tile_d1, stride0);
  v4i z4 = {0, 0, 0, 0};
#if defined(__clang_major__) && (__clang_major__ >= 23)
  v8i z8 = {0, 0, 0, 0, 0, 0, 0, 0};
  __builtin_amdgcn_tensor_load_to_lds(g0, g1, z4, z4, z8, CPOL_NT);
#else
  __builtin_amdgcn_tensor_load_to_lds(g0, g1, z4, z4, CPOL_NT);
#endif
}

__device__ __forceinline__ void tdm_store_2d_f32(
    unsigned lds_byte_off, unsigned long long gaddr,
    unsigned tensor_d0, unsigned tensor_d1,
    unsigned tile_d0, unsigned tile_d1, unsigned stride0)
{
  v4u g0; v8i g1;
  tdm_build_2d_f32(g0, g1, lds_byte_off, gaddr, tensor_d0, tensor_d1,
                   tile_d0, tile_d1, stride0);
  v4i z4 = {0, 0, 0, 0};
#if defined(__clang_major__) && (__clang_major__ >= 23)
  v8i z8 = {0, 0, 0, 0, 0, 0, 0, 0};
  __builtin_amdgcn_tensor_store_from_lds(g0, g1, z4, z4, z8, CPOL_NT);
#else
  __builtin_amdgcn_tensor_store_from_lds(g0, g1, z4, z4, CPOL_NT);
#endif
}

__device__ __forceinline__ unsigned lds_off_of(const void* p)
{
  // addrspace(3) pointers live in the LDS aperture; low 32 bits are the byte
  // offset within the wave's LDS allocation (ISA §10.2 aperture mapping).
  return (unsigned)(unsigned long long)(uintptr_t)p;
}
#endif // USE_TDM

// ---------------------------------------------------------------------------
// Kernel: one block handles ROWS rows. TDM stages in[63]/out[483] tiles in LDS.
// ---------------------------------------------------------------------------
__global__ __launch_bounds__(NT)
void angular_layer_kernel(const float* __restrict__ in,
                          float* __restrict__ out, int B)
{
  __shared__ float s_in[ROWS * IN_COLS];          //  4 032 B
  __shared__ float s_out[ROWS * OUT_COLS];        // 30 912 B
  __shared__ unsigned short s_pair[NPAIRS];       // i | (k<<8)

  const int tid  = threadIdx.x;
  const int row0 = blockIdx.x * ROWS;
  if (row0 >= B) return;                          // uniform per block
  const int rows = (B - row0 < ROWS) ? (B - row0) : ROWS;

  // Build (i,k) pair LUT once per block.
  if (tid == 0) {
    int idx = 0;
    for (int i = 0; i < 20; ++i)
      for (int k = i + 1; k < 21; ++k)
        s_pair[idx++] = (unsigned short)(i | (k << 8));
  }

  // ---- Stage input tile [rows x 63] into LDS --------------------------------
#if USE_TDM
  if ((tid >> 5) == 0) {                          // wave 0 only (uniform branch)
    tdm_load_2d_f32(lds_off_of(s_in),
                    (unsigned long long)(uintptr_t)(in + (size_t)row0 * IN_COLS),
                    /*tensor_d0=*/IN_COLS, /*tensor_d1=*/(unsigned)(B - row0),
                    /*tile_d0=*/IN_COLS,   /*tile_d1=*/(unsigned)rows,
                    /*stride0=*/IN_COLS);
    __builtin_amdgcn_s_wait_tensorcnt(0);
  }
#else
  for (int f = tid; f < rows * IN_COLS; f += NT)
    s_in[f] = in[(size_t)row0 * IN_COLS + f];
#endif
  __syncthreads();

  // ---- Pass-through columns 0..62 ------------------------------------------
  for (int f = tid; f < rows * IN_COLS; f += NT) {
    int r = f / IN_COLS;
    int c = f - r * IN_COLS;
    s_out[r * OUT_COLS + c] = s_in[f];
  }

  // ---- 210 pair tilts per row ----------------------------------------------
  const int ntask = rows * NPAIRS;
  for (int t = tid; t < ntask; t += NT) {
    int r = t / NPAIRS;
    int p = t - r * NPAIRS;
    unsigned pk = s_pair[p];
    int i = (int)(pk & 0xFFu);
    int k = (int)(pk >> 8);
    const float* pr = &s_in[r * IN_COLS];
    float dx = pr[3 * k]     - pr[3 * i];
    float dy = pr[3 * k + 1] - pr[3 * i + 1];
    float inv = 1.0f / __builtin_sqrtf(dx * dx + dy * dy);
    float tx = dx * inv;
    float ty = dy * inv;
    tx = fminf(fmaxf(tx, -1.0f), 1.0f);
    ty = fminf(fmaxf(ty, -1.0f), 1.0f);
    float* po = &s_out[r * OUT_COLS + IN_COLS + 2 * p];
    po[0] = tx;
    po[1] = ty;
  }
  __syncthreads();

  // ---- Stream output tile [rows x 483] to HBM (dominant traffic) -----------
#if USE_TDM
  if ((tid >> 5) == 0) {                          // wave 0 only
    tdm_store_2d_f32(lds_off_of(s_out),
                     (unsigned long long)(uintptr_t)(out + (size_t)row0 * OUT_COLS),
                     /*tensor_d0=*/OUT_COLS, /*tensor_d1=*/(unsigned)(B - row0),
                     /*tile_d0=*/OUT_COLS,   /*tile_d1=*/(unsigned)rows,
                     /*stride0=*/OUT_COLS);
    __builtin_amdgcn_s_wait_tensorcnt(0);
  }
#else
  for (int f = tid; f < rows * OUT_COLS; f += NT)
    out[(size_t)row0 * OUT_COLS + f] = s_out[f];
#endif
}

// ---------------------------------------------------------------------------
extern "C" void kernel_launch(void* const* d_in, const int* in_sizes, int n_in,
                              void* d_out, int out_size, void* d_ws, size_t ws_size,
                              hipStream_t stream)
{
  (void)n_in; (void)out_size; (void)d_ws; (void)ws_size;
  const float* in  = (const float*)d_in[0];
  float*       out = (float*)d_out;
  const int B = in_sizes[0] / IN_COLS;            // 500000
  const int grid = (B + ROWS - 1) / ROWS;         // 31250 blocks
  angular_layer_kernel<<<grid, NT, 0, stream>>>(in, out, B);
}